// MinSumBPDecoder_65592740544939
// MI455X (gfx1250) — compile-verified
//
#include <hip/hip_runtime.h>
#include <hip/hip_bf16.h>
#include <stdint.h>

// MinSum LDPC BP decoder for MI455X (gfx1250).
// Messages live in ONE in-place buffer buf[(e,b)] (E rows x 64 batch),
// batch-contiguous: a row is 256B = one L2 line. Whole 16MB state is
// L2-resident (192MB L2) => L2-BW bound; variable-side kernels use float2
// (2 batch elems/lane) so one wave32 vmem op moves a full 256B row.
// Check-update stages rows into LDS with gfx1250 async-to-LDS (ASYNCcnt).
// All hot-path indexing is 32-bit (values < 2^23) to avoid v_mul_u64 chains.

#define MM 8192        // checks
#define NN 16384       // variables
#define BB 64          // batch
#define N_ITER 12
#define D_ALPHA 0.8f
#define D_CLAMP 20.0f
#define D_PADBIG 1000000.0f
#define D_TOL 1e-9f
#define MAXCD 8        // check degree cap (DC=8, collisions only reduce it)
#define MAXVD 4        // var degree cap (DV=4)
#define CPB 4          // checks per 256-thread block in check kernels

#if defined(__AMDGCN__) && defined(__gfx1250__)
# if __has_builtin(__builtin_amdgcn_global_load_async_to_lds_b32)
#  define HAVE_ASYNC_LDS 1
# endif
#endif

__device__ __forceinline__ void stage_to_lds(float* lds_dst, const float* g_src) {
#ifdef HAVE_ASYNC_LDS
  // Prototype (probe-verified): (global int*, shared int*, imm offset, imm cpol)
  __builtin_amdgcn_global_load_async_to_lds_b32(
      (__attribute__((address_space(1))) int*)(int*)(void*)g_src,
      (__attribute__((address_space(3))) int*)(int*)(void*)lds_dst,
      0, 0);
#else
  *lds_dst = *g_src;
#endif
}

__device__ __forceinline__ void wait_stage() {
#ifdef HAVE_ASYNC_LDS
# if __has_builtin(__builtin_amdgcn_s_wait_asynccnt)
  __builtin_amdgcn_s_wait_asynccnt(0);
# else
  asm volatile("s_wait_asynccnt 0" ::: "memory");
# endif
#endif
}

// ------------- check-node update (in-place vtc -> ctv) -------------------
// Defined FIRST so the disasm snippet shows the async-to-LDS path.
__global__ void k_chk_update(const float* __restrict__ sT,
                             const int* __restrict__ chk_adj,
                             const float* __restrict__ chk_mask,
                             float* __restrict__ buf, int max_cd) {
  __shared__ float stage[CPB * MAXCD * BB];  // 8 KB
  int tid = (int)threadIdx.x;
  int b = tid & (BB - 1);
  int grp = tid >> 6;
  int c = (int)blockIdx.x * CPB + grp;
  int cbase = c * max_cd;

  float msk[MAXCD];
  int ei[MAXCD];
#pragma unroll
  for (int s = 0; s < MAXCD; ++s) {
    bool ok = (s < max_cd) && (chk_mask[cbase + s] != 0.0f);
    msk[s] = ok ? 1.0f : 0.0f;
    ei[s] = ok ? chk_adj[cbase + s] : 0;
    if (ok)
      stage_to_lds(&stage[(grp * MAXCD + s) * BB + b], &buf[ei[s] * BB + b]);
  }
  wait_stage();  // s_wait_asynccnt 0 (per-wave; each thread reads only its own slots)

  float sprod = sT[c * BB + b];  // s_sign = 1-2*syndrome
  float m1 = 1e30f, m2 = 1e30f;
  float sg[MAXCD], av[MAXCD];
#pragma unroll
  for (int s = 0; s < MAXCD; ++s) {
    bool ok = (msk[s] != 0.0f);
    float g = ok ? stage[(grp * MAXCD + s) * BB + b] : 0.0f;
    float s_ = ok ? ((g < 0.0f) ? -1.0f : 1.0f) : 1.0f;  // sign(0)->+1; pad->+1
    float a = ok ? fabsf(g) : D_PADBIG;                  // pad slots ~1e6
    sg[s] = s_;
    av[s] = a;
    sprod *= s_;
    if (a < m1) { m2 = m1; m1 = a; } else if (a < m2) { m2 = a; }
  }
#pragma unroll
  for (int s = 0; s < MAXCD; ++s) {
    if (msk[s] != 0.0f) {
      float excl = (fabsf(av[s] - m1) < D_TOL) ? m2 : m1;
      buf[ei[s] * BB + b] = D_ALPHA * sprod * sg[s] * excl;
    }
  }
}

// ---------------- zero init of message buffer ----------------
__global__ void k_zero(float4* __restrict__ p, int n4) {
  int i = (int)blockIdx.x * (int)blockDim.x + (int)threadIdx.x;
  if (i < n4) p[i] = make_float4(0.f, 0.f, 0.f, 0.f);
}

// ------------- transpose (B=64, C) -> (C, 64); mode1: x -> 1-2x ------------
__global__ void k_tr_in(const float* __restrict__ in, float* __restrict__ out,
                        int C, int mode) {
  __shared__ float t[32][33];
  int c0 = (int)blockIdx.x * 32, r0 = (int)blockIdx.y * 32;
  int tx = (int)threadIdx.x, ty = (int)threadIdx.y;
#pragma unroll
  for (int j = 0; j < 32; j += 8) {
    float v = in[(r0 + ty + j) * C + (c0 + tx)];
    t[ty + j][tx] = mode ? (1.0f - 2.0f * v) : v;
  }
  __syncthreads();
#pragma unroll
  for (int j = 0; j < 32; j += 8)
    out[(c0 + ty + j) * BB + (r0 + tx)] = t[tx][ty + j];
}

// ------------- variable-node update (in-place ctv -> vtc) ----------------
// float2: each lane handles 2 batch elems; wave32 covers a full 256B row.
__global__ void k_var_update(const float* __restrict__ llrT,
                             const int* __restrict__ var_adj,
                             const float* __restrict__ var_mask,
                             float* __restrict__ buf, int max_vd) {
  int tid = (int)blockIdx.x * 256 + (int)threadIdx.x;
  int b2 = tid & 31;
  int v = tid >> 5;
  int vbase = v * max_vd;
  const float2* llrT2 = (const float2*)llrT;
  float2* buf2 = (float2*)buf;
  float2 llr = llrT2[v * 32 + b2];
  float2 m[MAXVD];
  float mk[MAXVD];
  int e[MAXVD];
  float2 sum = make_float2(0.f, 0.f);
#pragma unroll
  for (int s = 0; s < MAXVD; ++s) {
    bool ok = (s < max_vd) && (var_mask[vbase + s] != 0.0f);
    mk[s] = ok ? 1.0f : 0.0f;
    e[s] = ok ? var_adj[vbase + s] : 0;
    m[s] = ok ? buf2[e[s] * 32 + b2] : make_float2(0.f, 0.f);
    sum.x += m[s].x;
    sum.y += m[s].y;
  }
  float2 vt = make_float2(llr.x + sum.x, llr.y + sum.y);
#pragma unroll
  for (int s = 0; s < MAXVD; ++s) {
    if (mk[s] != 0.0f) {
      float2 x;
      x.x = fminf(fmaxf(vt.x - m[s].x, -D_CLAMP), D_CLAMP);
      x.y = fminf(fmaxf(vt.y - m[s].y, -D_CLAMP), D_CLAMP);
      buf2[e[s] * 32 + b2] = x;
    }
  }
}

// ------------- final marginals (N,64 layout), float2 ----------------------
__global__ void k_marginals(const float* __restrict__ llrT,
                            const int* __restrict__ var_adj,
                            const float* __restrict__ var_mask,
                            const float* __restrict__ buf,
                            float* __restrict__ margT, int max_vd) {
  int tid = (int)blockIdx.x * 256 + (int)threadIdx.x;
  int b2 = tid & 31;
  int v = tid >> 5;
  int vbase = v * max_vd;
  const float2* llrT2 = (const float2*)llrT;
  const float2* buf2 = (const float2*)buf;
  float2* margT2 = (float2*)margT;
  float2 total = llrT2[v * 32 + b2];
#pragma unroll
  for (int s = 0; s < MAXVD; ++s) {
    bool ok = (s < max_vd) && (var_mask[vbase + s] != 0.0f);
    if (ok) {
      float2 m = buf2[var_adj[vbase + s] * 32 + b2];
      total.x += m.x;
      total.y += m.y;
    }
  }
  float2 r;
  r.x = 1.0f / (1.0f + expf(total.x));  // sigmoid(-total)
  r.y = 1.0f / (1.0f + expf(total.y));
  margT2[v * 32 + b2] = r;
}

// ------------- (N,64) -> (64,N) output transpose; also hard decisions ----
__global__ void k_out_tr(const float* __restrict__ margT,
                         float* __restrict__ outM, float* __restrict__ outH) {
  __shared__ float t[32][33];
  int v0 = (int)blockIdx.x * 32, b0 = (int)blockIdx.y * 32;
  int tx = (int)threadIdx.x, ty = (int)threadIdx.y;
#pragma unroll
  for (int j = 0; j < 32; j += 8)
    t[ty + j][tx] = margT[(v0 + ty + j) * BB + (b0 + tx)];
  __syncthreads();
#pragma unroll
  for (int j = 0; j < 32; j += 8) {
    float m = t[tx][ty + j];
    int o = (b0 + ty + j) * NN + (v0 + tx);
    outM[o] = m;
    outH[o] = (m > 0.5f) ? 1.0f : 0.0f;
  }
}

__global__ void k_conv_init(float* __restrict__ conv) {
  conv[threadIdx.x] = 1.0f;
}

// ------------- parity check / convergence --------------------------------
__global__ void k_parity(const float* __restrict__ sT,
                         const int* __restrict__ chk_adj,
                         const float* __restrict__ chk_mask,
                         const int* __restrict__ var_idx,
                         const float* __restrict__ margT,
                         float* __restrict__ conv, int max_cd) {
  int tid = (int)threadIdx.x;
  int b = tid & (BB - 1);
  int grp = tid >> 6;
  int c = (int)blockIdx.x * CPB + grp;
  int cbase = c * max_cd;
  int sum = 0;
#pragma unroll
  for (int s = 0; s < MAXCD; ++s) {
    bool ok = (s < max_cd) && (chk_mask[cbase + s] != 0.0f);
    if (ok) {
      int v = var_idx[chk_adj[cbase + s]];
      sum += (margT[v * BB + b] > 0.5f) ? 1 : 0;
    }
  }
  int syn = (sT[c * BB + b] < 0.0f) ? 1 : 0;
  if ((sum & 1) != syn) conv[b] = 0.0f;  // race-free: all writers store 0
}

extern "C" void kernel_launch(void* const* d_in, const int* in_sizes, int n_in,
                              void* d_out, int out_size, void* d_ws, size_t ws_size,
                              hipStream_t stream) {
  const float* syndrome  = (const float*)d_in[0];   // (64, 8192)
  const float* llr       = (const float*)d_in[1];   // (64, 16384)
  const int*   var_idx   = (const int*)d_in[3];     // (E,)
  const int*   check_adj = (const int*)d_in[4];     // (M, max_cd)
  const float* check_mask= (const float*)d_in[5];
  const int*   var_adj   = (const int*)d_in[6];     // (N, max_vd)
  const float* var_mask  = (const float*)d_in[7];

  const int E      = in_sizes[8];           // inv_perm length == edge count
  const int max_cd = in_sizes[4] / MM;
  const int max_vd = in_sizes[6] / NN;

  // Workspace: buf (E*64) | llrT (N*64) | sT (M*64) | margT (N*64)  ~27 MB
  float* buf   = (float*)d_ws;
  float* llrT  = buf  + (size_t)E * BB;
  float* sT    = llrT + (size_t)NN * BB;
  float* margT = sT   + (size_t)MM * BB;

  float* outM = (float*)d_out;                       // marginals (64, N)
  float* outH = outM + (size_t)BB * NN;              // hard decisions (64, N)
  float* conv = outH + (size_t)BB * NN;              // converged (64,)

  // init: ctv = 0; transpose inputs to batch-contiguous layout
  {
    int n4 = (E * BB) / 4;
    k_zero<<<(n4 + 255) / 256, 256, 0, stream>>>((float4*)buf, n4);
  }
  k_tr_in<<<dim3(NN / 32, 2), dim3(32, 8), 0, stream>>>(llr, llrT, NN, 0);
  k_tr_in<<<dim3(MM / 32, 2), dim3(32, 8), 0, stream>>>(syndrome, sT, MM, 1);

  for (int it = 0; it < N_ITER; ++it) {
    k_var_update<<<(NN * 32) / 256, 256, 0, stream>>>(llrT, var_adj, var_mask,
                                                      buf, max_vd);
    k_chk_update<<<MM / CPB, 256, 0, stream>>>(sT, check_adj, check_mask,
                                               buf, max_cd);
  }

  k_marginals<<<(NN * 32) / 256, 256, 0, stream>>>(llrT, var_adj, var_mask,
                                                   buf, margT, max_vd);
  k_out_tr<<<dim3(NN / 32, 2), dim3(32, 8), 0, stream>>>(margT, outM, outH);
  k_conv_init<<<1, BB, 0, stream>>>(conv);
  k_parity<<<MM / CPB, 256, 0, stream>>>(sT, check_adj, check_mask, var_idx,
                                         margT, conv, max_cd);
}